// HungarianMatcher_36429912605482
// MI455X (gfx1250) — compile-verified
//
#include <hip/hip_runtime.h>
#include <hip/hip_bf16.h>
#include <float.h>

#define BS 16
#define NQ 900
#define NC 80
#define NT 128
#define NROWS (BS * NQ)   // 14400
#define NCOLS (BS * NT)   // 2048
#define RPB 8             // rows per block in main kernel
#define THREADS 256

#define COST_CLASS_W 2.0f
#define COST_BBOX_W 5.0f
#define COST_GIOU_W 2.0f
#define EPS_F 1e-8f

// ---- CDNA5 async global->LDS path (ASYNCcnt), guarded so we always compile ----
#if defined(__AMDGCN__) && __has_builtin(__builtin_amdgcn_global_load_async_to_lds_b128)
#define USE_ASYNC 1
#else
#define USE_ASYNC 0
#endif

__device__ __forceinline__ void wait_async_zero() {
#if defined(__AMDGCN__)
#if __has_builtin(__builtin_amdgcn_s_wait_asynccnt)
  __builtin_amdgcn_s_wait_asynccnt(0);
#else
  asm volatile("s_wait_asynccnt 0" ::: "memory");
#endif
#endif
}

#if USE_ASYNC
typedef __attribute__((ext_vector_type(4))) int v4i_t;
typedef __attribute__((address_space(1))) v4i_t* g_v4i_p;
typedef __attribute__((address_space(3))) v4i_t* l_v4i_p;

__device__ __forceinline__ void async_cp16(void* lds_dst, const void* gmem_src) {
  __builtin_amdgcn_global_load_async_to_lds_b128(
      (g_v4i_p)(v4i_t*)(void*)gmem_src,
      (l_v4i_p)(v4i_t*)lds_dst,
      0, 0);
}
#endif

// ---------------- Kernel 3 (emitted first for disasm visibility) ----------------
// Main cost matrix (14400 x 2048):
//   C = 5*L1(cxcywh) + cc[row, label[col]] - 2*GIoU, then nan_to_num semantics.
__global__ __launch_bounds__(THREADS) void matcher_main_kernel(
    const float* __restrict__ pred_boxes, const float* __restrict__ cc,
    const float* __restrict__ tplanes, float* __restrict__ out) {
  // 16B alignment required: async b128 writes LDS in 16-byte units.
  __shared__ __align__(16) float s_t[6 * NCOLS];   // 48 KB target planes
  __shared__ __align__(16) float s_cc[RPB * NC];   // 2.5 KB focal-cost rows

  const int tid = threadIdx.x;
  const int rowBase = blockIdx.x * RPB;

#if USE_ASYNC
  {
    const float4* gt = (const float4*)tplanes;
    float4* st = (float4*)s_t;
    for (int i = tid; i < (6 * NCOLS) / 4; i += THREADS)
      async_cp16(&st[i], &gt[i]);
    const float4* gc = (const float4*)(cc + rowBase * NC);  // rowBase*320B: 16B aligned
    float4* sc = (float4*)s_cc;
    for (int i = tid; i < (RPB * NC) / 4; i += THREADS)
      async_cp16(&sc[i], &gc[i]);
    wait_async_zero();
  }
#else
  for (int i = tid; i < 6 * NCOLS; i += THREADS) s_t[i] = tplanes[i];
  for (int i = tid; i < RPB * NC; i += THREADS) s_cc[i] = cc[rowBase * NC + i];
#endif
  __syncthreads();

  // Per-row pred-box data held in registers (8 rows x 9 values).
  float pcx[RPB], pcy[RPB], pw[RPB], ph[RPB];
  float px1[RPB], py1[RPB], px2[RPB], py2[RPB], pa[RPB];
#pragma unroll
  for (int r = 0; r < RPB; ++r) {
    const float4 b = ((const float4*)pred_boxes)[rowBase + r];
    pcx[r] = b.x; pcy[r] = b.y; pw[r] = b.z; ph[r] = b.w;
    px1[r] = b.x - 0.5f * b.z;
    py1[r] = b.y - 0.5f * b.w;
    px2[r] = b.x + 0.5f * b.z;
    py2[r] = b.y + 0.5f * b.w;
    pa[r] = (px2[r] - px1[r]) * (py2[r] - py1[r]);
  }

  const float* t_cx = s_t;
  const float* t_cy = s_t + NCOLS;
  const float* t_w  = s_t + 2 * NCOLS;
  const float* t_h  = s_t + 3 * NCOLS;
  const float* t_ar = s_t + 4 * NCOLS;
  const int*   t_lb = (const int*)(s_t + 5 * NCOLS);

  for (int c = tid; c < NCOLS; c += THREADS) {  // lanes stride cols: coalesced stores
    const float tcx = t_cx[c], tcy = t_cy[c], tw = t_w[c], th = t_h[c];
    const float ta = t_ar[c];
    const int lb = t_lb[c];
    const float tx1 = tcx - 0.5f * tw, ty1 = tcy - 0.5f * th;
    const float tx2 = tcx + 0.5f * tw, ty2 = tcy + 0.5f * th;
#pragma unroll
    for (int r = 0; r < RPB; ++r) {
      const float ltx = fmaxf(px1[r], tx1), lty = fmaxf(py1[r], ty1);
      const float rbx = fminf(px2[r], tx2), rby = fminf(py2[r], ty2);
      const float inter = fmaxf(rbx - ltx, 0.0f) * fmaxf(rby - lty, 0.0f);
      const float uni = pa[r] + ta - inter;
      const float iou = inter / uni;
      const float cx1 = fminf(px1[r], tx1), cy1 = fminf(py1[r], ty1);
      const float cx2 = fmaxf(px2[r], tx2), cy2 = fmaxf(py2[r], ty2);
      const float ca = fmaxf(cx2 - cx1, 0.0f) * fmaxf(cy2 - cy1, 0.0f);
      const float giou = iou - (ca - uni) / ca;
      const float l1 = fabsf(pcx[r] - tcx) + fabsf(pcy[r] - tcy) +
                       fabsf(pw[r] - tw) + fabsf(ph[r] - th);
      float cost = COST_BBOX_W * l1 + s_cc[r * NC + lb] - COST_GIOU_W * giou;
      // jnp.nan_to_num(C, nan=1.0): NaN->1.0, +inf->FLT_MAX, -inf->-FLT_MAX
      if (!(cost == cost)) cost = 1.0f;
      else if (cost > FLT_MAX) cost = FLT_MAX;
      else if (cost < -FLT_MAX) cost = -FLT_MAX;
      // Streamed once, never re-read by the kernel: non-temporal store.
      __builtin_nontemporal_store(cost, &out[(size_t)(rowBase + r) * NCOLS + c]);
    }
  }
}

// ---------------- Kernel 1: focal class-cost table (14400 x 80) ----------------
// cc[q, c] = COST_CLASS * (pos - neg) at p = sigmoid(logits[q, c])
__global__ __launch_bounds__(THREADS) void class_cost_kernel(
    const float* __restrict__ logits, float* __restrict__ cc) {
  const int i = blockIdx.x * THREADS + threadIdx.x;  // grid exactly covers NROWS*NC
  const float x = logits[i];
  const float p = 1.0f / (1.0f + expf(-x));
  const float neg = 0.75f * p * p * (-logf(1.0f - p + EPS_F));
  const float pos = 0.25f * (1.0f - p) * (1.0f - p) * (-logf(p + EPS_F));
  cc[i] = COST_CLASS_W * (pos - neg);
}

// ---------------- Kernel 2: target SoA planes (6 x 2048) ----------------
// planes: [0]=cx [1]=cy [2]=w [3]=h [4]=area(from xyxy) [5]=label(int bits)
__global__ __launch_bounds__(THREADS) void prep_targets_kernel(
    const int* __restrict__ labels_raw, const float* __restrict__ tboxes,
    float* __restrict__ tplanes) {
  __shared__ int s_nonzero;
  const int tid = threadIdx.x;
  if (tid == 0) s_nonzero = 0;
  __syncthreads();
  // int64-vs-int32 detection: if labels were int64, the high dwords (odd
  // indices) of the first 1024 values are all zero. Bounds stay within the
  // minimal (int32) buffer size of 2048 dwords.
  for (int t = tid; t < 1024; t += THREADS)
    if (labels_raw[2 * t + 1] != 0) s_nonzero = 1;
  __syncthreads();
  const bool is64 = (s_nonzero == 0);
  for (int t = tid; t < NCOLS; t += THREADS) {
    const int lb = is64 ? labels_raw[2 * t] : labels_raw[t];
    const float4 b = ((const float4*)tboxes)[t];  // cx, cy, w, h
    const float x1 = b.x - 0.5f * b.z, y1 = b.y - 0.5f * b.w;
    const float x2 = b.x + 0.5f * b.z, y2 = b.y + 0.5f * b.w;
    tplanes[0 * NCOLS + t] = b.x;
    tplanes[1 * NCOLS + t] = b.y;
    tplanes[2 * NCOLS + t] = b.z;
    tplanes[3 * NCOLS + t] = b.w;
    tplanes[4 * NCOLS + t] = (x2 - x1) * (y2 - y1);
    ((int*)tplanes)[5 * NCOLS + t] = lb;
  }
}

extern "C" void kernel_launch(void* const* d_in, const int* in_sizes, int n_in,
                              void* d_out, int out_size, void* d_ws, size_t ws_size,
                              hipStream_t stream) {
  (void)in_sizes; (void)n_in; (void)out_size; (void)ws_size;
  const float* pred_logits = (const float*)d_in[0];
  const float* pred_boxes  = (const float*)d_in[1];
  const int*   tgt_labels  = (const int*)d_in[2];
  const float* tgt_boxes   = (const float*)d_in[3];
  float* out = (float*)d_out;

  // Workspace layout: [cc table: NROWS*NC f32][target planes: 6*NCOLS f32] ~4.7 MB
  float* cc = (float*)d_ws;
  float* tplanes = cc + (size_t)NROWS * NC;

  class_cost_kernel<<<(NROWS * NC) / THREADS, THREADS, 0, stream>>>(pred_logits, cc);
  prep_targets_kernel<<<1, THREADS, 0, stream>>>(tgt_labels, tgt_boxes, tplanes);
  matcher_main_kernel<<<NROWS / RPB, THREADS, 0, stream>>>(pred_boxes, cc, tplanes, out);
}